// DFT_Layer_74990128988640
// MI455X (gfx1250) — compile-verified
//
#include <hip/hip_runtime.h>

// Problem constants from the reference: B=1024, S=200, D=128.
#define SSEQ 200
#define DEMB 128

typedef __attribute__((ext_vector_type(2))) float v2f;
typedef __attribute__((ext_vector_type(8))) float v8f;

// One block (256 threads = 8 wave32) per batch row b.
// Phase 1: compute rolled/normalized DFT-magnitude weights w[0..S-1] in LDS.
// Phase 2: out[b, :] = w . E[b]  via chained v_wmma_f32_16x16x4_f32
//          (A rows all broadcast w[4t+kk] -> every D row equals the answer).
__global__ __launch_bounds__(256) void dft_pool_kernel(
    const int* __restrict__ seqs,       // [B, S] int32
    const float* __restrict__ emb,      // [B, S, D] f32
    float* __restrict__ out) {          // [B, D] f32
  __shared__ int   sh_seq[SSEQ];
  __shared__ short sh_occ[SSEQ];
  __shared__ float sh_mag[SSEQ];
  __shared__ float sh_w[SSEQ];
  __shared__ int   sh_L;
  __shared__ int   sh_nocc;
  __shared__ float sh_inv_total;

  const int b   = blockIdx.x;
  const int tid = threadIdx.x;

  if (tid == 0) sh_L = 0;
  __syncthreads();

  // Load tokens; count real length L (padding tokens are 0).
  if (tid < SSEQ) {
    int v = seqs[(size_t)b * SSEQ + tid];
    sh_seq[tid] = v;
    if (v != 0) atomicAdd(&sh_L, 1);
  }
  __syncthreads();

  // Deterministic serial scan for occurrence positions of the last real token.
  // (seqs[n]==last implies n<L since padding is 0 and last>=1.)
  if (tid == 0) {
    int L    = sh_L;
    int last = sh_seq[L - 1];
    int m = 0;
    for (int n = 0; n < SSEQ; ++n)
      if (sh_seq[n] == last) sh_occ[m++] = (short)n;
    sh_nocc = m;
  }
  __syncthreads();

  const int L = sh_L;
  const int M = sh_nocc;

  // |DFT_k| of the sparse binary indicator: sum over occurrence positions.
  if (tid < SSEQ) {
    float re = 0.f, im = 0.f;
    const float coef = -6.283185307179586f * (float)tid / (float)L;
    for (int j = 0; j < M; ++j) {
      float th = coef * (float)sh_occ[j];
      re += cosf(th);
      im += sinf(th);
    }
    float mag = sqrtf(re * re + im * im);
    if (tid >= L - 1) mag = 0.f;    // zero bin L-1 and all padding bins
    sh_mag[tid] = mag;
  }
  __syncthreads();

  // Deterministic serial normalization sum (tiny: 200 adds).
  if (tid == 0) {
    float tot = 0.f;
    for (int k = 0; k < SSEQ; ++k) tot += sh_mag[k];
    sh_inv_total = 1.0f / tot;
  }
  __syncthreads();

  // Normalize + roll right by one with zero fill.
  if (tid < SSEQ) sh_w[tid] = (tid == 0) ? 0.f : sh_mag[tid - 1] * sh_inv_total;
  __syncthreads();

  // ---- Phase 2: WMMA contraction, one wave per 16-wide d-tile ----
  const int wave = tid >> 5;          // 0..7
  const int lane = tid & 31;
  const int hi   = lane >> 4;         // 0: lanes 0-15, 1: lanes 16-31
  const int nl   = lane & 15;
  const int n0   = wave << 4;         // d-tile base
  const float* Eb = emb + (size_t)b * (SSEQ * DEMB);

  v8f acc = {};
  #pragma unroll 2
  for (int t = 0; t < SSEQ / 4; ++t) {
    const int r = 4 * t + 2 * hi;     // s-row pair handled by this lane half
    // Unconditional speculative prefetch ~6 quads (12 KB) ahead into GL2.
    // Past-the-end addresses on the final blocks are silently dropped
    // (speculative SYS-scope prefetch); 105 MB working set fits in 192 MB L2.
    __builtin_prefetch(Eb + (size_t)(4 * (t + 6) + 2 * hi) * DEMB + n0 + nl, 0, 0);
    // A (16x4 f32): VGPR0 = K{0|2}, VGPR1 = K{1|3}; identical for every row m.
    v2f a, bv;
    a[0] = sh_w[r];
    a[1] = sh_w[r + 1];
    // B (4x16 f32): VGPR0 = row K{0|2}, VGPR1 = row K{1|3}, N = lane%16.
    bv[0] = Eb[(size_t)r * DEMB + n0 + nl];
    bv[1] = Eb[(size_t)(r + 1) * DEMB + n0 + nl];
    // D[m,n] += sum_kk w[4t+kk] * E[4t+kk, n0+n]  (same for all m)
    acc = __builtin_amdgcn_wmma_f32_16x16x4_f32(
        /*neg_a=*/false, a, /*neg_b=*/false, bv,
        /*c_mod=*/(short)0, acc, /*reuse_a=*/false, /*reuse_b=*/false);
  }

  // Row M=0 of D lives in acc[0] of lanes 0-15 (N = lane).
  if (lane < 16) out[(size_t)b * DEMB + n0 + lane] = acc[0];
}

extern "C" void kernel_launch(void* const* d_in, const int* in_sizes, int n_in,
                              void* d_out, int out_size, void* d_ws, size_t ws_size,
                              hipStream_t stream) {
  const int*   seqs = (const int*)d_in[0];          // [B,S] int32
  const float* emb  = (const float*)d_in[1];        // [B,S,D] f32
  float*       out  = (float*)d_out;                // [B,D] f32
  const int B = in_sizes[0] / SSEQ;
  hipLaunchKernelGGL(dft_pool_kernel, dim3(B), dim3(256), 0, stream, seqs, emb, out);
}